// MultiLoss_20641612824937
// MI455X (gfx1250) — compile-verified
//
#include <hip/hip_runtime.h>
#include <stdint.h>

// ---------------------------------------------------------------------------
// MultiBox loss for MI455X (gfx1250, wave32).
// Memory-bound: ~248MB read once -> ~11us floor at 23.3 TB/s. No matmul, so
// no WMMA; CDNA5-specific paths used instead:
//   * TDM tensor_load_to_lds (+ s_wait_tensorcnt) to bulk-copy a whole
//      256KB anchor row into LDS (only possible with CDNA5's 320KB/WGP LDS)
//   * wave32 __shfl_xor reductions, LDS atomics, global_prefetch_b8
// Top-K-sum is tie-invariant, so hard-negative mining reduces to a radix
// select over order-preserving float keys instead of a double argsort.
// ---------------------------------------------------------------------------

#define N_IMG   32
#define A_ANCH  65536
#define C_CLS   21
#define NEG_POS 4
#define K2_THREADS 1024

typedef uint32_t u32x4 __attribute__((ext_vector_type(4)));
typedef int      i32x8 __attribute__((ext_vector_type(8)));
typedef int      i32x4 __attribute__((ext_vector_type(4)));

__device__ __forceinline__ uint32_t f2key(float f) {
    uint32_t u = __float_as_uint(f);
    return (u & 0x80000000u) ? ~u : (u | 0x80000000u);   // order-preserving
}
__device__ __forceinline__ float key2f(uint32_t k) {
    uint32_t u = (k & 0x80000000u) ? (k & 0x7FFFFFFFu) : ~k;
    return __uint_as_float(u);
}

// ---------------------------------------------------------------------------
// Kernel 0: zero accumulators.
__global__ void k_init(int* __restrict__ numpos, float* __restrict__ acc) {
    int t = threadIdx.x;
    if (t < N_IMG) numpos[t] = 0;
    if (t < 2)     acc[t]    = 0.0f;   // acc[0]=loc_sum, acc[1]=conf_sum
}

// ---------------------------------------------------------------------------
// Kernel 1: per-anchor CE + smooth-L1; writes loss_c; accumulates sums.
// One thread per anchor; each 256-thread block lies entirely inside one row
// (A = 65536 is a multiple of 256), so num_pos accumulation is per-row.
__global__ __launch_bounds__(256)
void k_ce(const float* __restrict__ regs, const float* __restrict__ cls,
          const float* __restrict__ coords, const int* __restrict__ labels,
          float* __restrict__ lossc, int* __restrict__ numpos,
          float* __restrict__ acc) {
    const int i = blockIdx.x * 256 + threadIdx.x;   // anchor id in [0, N*A)
    const int n = i >> 16;                          // row (A = 2^16)

    // Prefetch the next block's classification data (global_prefetch_b8).
    if (i + 256 < N_IMG * A_ANCH)
        __builtin_prefetch(cls + (size_t)(i + 256) * C_CLS, 0, 1);

    const int  label = labels[i];
    const bool pos   = label > 0;

    // Stable log-softmax CE over 21 classes.
    const float* __restrict__ x = cls + (size_t)i * C_CLS;
    float v[C_CLS];
    float m = -3.402823466e38f;
#pragma unroll
    for (int c = 0; c < C_CLS; ++c) { v[c] = x[c]; m = fmaxf(m, v[c]); }
    float s = 0.0f;
#pragma unroll
    for (int c = 0; c < C_CLS; ++c) s += __expf(v[c] - m);
    const float lse = m + __logf(s);
    const float ce  = lse - v[label];

    // Smooth-L1 (beta = 1) on positives.
    float locp = 0.0f;
    if (pos) {
#pragma unroll
        for (int k = 0; k < 4; ++k) {
            float d  = regs[(size_t)i * 4 + k] - coords[(size_t)i * 4 + k];
            float ad = fabsf(d);
            locp += (ad < 1.0f) ? 0.5f * d * d : (ad - 0.5f);
        }
    }

    lossc[i] = pos ? 0.0f : ce;          // hard-negative candidate value
    float confp = pos ? ce : 0.0f;       // positives always selected
    int   np    = pos ? 1 : 0;

    // wave32 butterfly reduction, then per-block LDS combine, then 3 atomics.
#pragma unroll
    for (int o = 16; o > 0; o >>= 1) {
        locp  += __shfl_xor(locp,  o, 32);
        confp += __shfl_xor(confp, o, 32);
        np    += __shfl_xor(np,    o, 32);
    }
    __shared__ float sl, sc;
    __shared__ int   sn;
    if (threadIdx.x == 0) { sl = 0.0f; sc = 0.0f; sn = 0; }
    __syncthreads();
    if ((threadIdx.x & 31) == 0) {
        atomicAdd(&sl, locp);            // ds_add_f32
        atomicAdd(&sc, confp);
        atomicAdd(&sn, np);
    }
    __syncthreads();
    if (threadIdx.x == 0) {
        atomicAdd(&acc[0], sl);
        atomicAdd(&acc[1], sc);
        atomicAdd(&numpos[n], sn);
    }
}

// ---------------------------------------------------------------------------
// Kernel 2: per-row top-K sum of loss_c via MSB-first radix select.
// One workgroup per row; the entire 256KB row lives in LDS (CDNA5: 320KB/WGP),
// loaded by the Tensor Data Mover as a 256x256 f32 tile.
__global__ __launch_bounds__(K2_THREADS)
void k_topk(const float* __restrict__ lossc, const int* __restrict__ numpos,
            float* __restrict__ acc) {
    __shared__ float    vals[A_ANCH];          // 256 KB
    __shared__ uint32_t hist[256];
    __shared__ uint32_t sh_prefix, sh_want;
    __shared__ float    redbuf[K2_THREADS / 32];

    const int n   = blockIdx.x;
    const int tid = threadIdx.x;

    const int np = numpos[n];
    int K = NEG_POS * np;
    if (K > A_ANCH - 1) K = A_ANCH - 1;
    if (K <= 0) return;                        // uniform: whole block exits

    const float* __restrict__ src = lossc + (size_t)n * A_ANCH;

#if defined(__has_builtin) && __has_builtin(__builtin_amdgcn_tensor_load_to_lds) && __has_builtin(__builtin_amdgcn_s_wait_tensorcnt)
    if (tid < 32) {                            // one wave issues the TDM op
        const uint32_t lds_off = (uint32_t)(uintptr_t)&vals[0]; // low 32b = LDS byte offset
        const uint64_t ga      = (uint64_t)(uintptr_t)src;
        // D# group 0: count=1 | lds_addr | global_addr[56:0] | type=2
        u32x4 g0 = { 1u, lds_off, (uint32_t)ga,
                     (uint32_t)(ga >> 32) | (2u << 30) };
        // D# group 1: data_size=4B; tensor 256x256 f32, tile 256x256,
        // dim0_stride=256 -> contiguous 64K-element row as a 2D tile
        // (tile_dim0 is only 16 bits, so 1x65536 is not encodable).
        i32x8 g1 = { (int)(2u << 16),          // data_size = 2 (4 bytes)
                     (int)(256u << 16),        // tensor_dim0 lo16 -> bits 63:48
                     (int)(256u << 16),        // dim0 hi16=0, tensor_dim1 lo16
                     (int)(256u << 16),        // dim1 hi16=0, tile_dim0 = 256
                     256,                      // tile_dim1 = 256, tile_dim2 = 0
                     256,                      // tensor_dim0_stride lo32 = 256
                     0, 0 };                   // stride hi bits / dim1_stride
        i32x4 gz4 = { 0, 0, 0, 0 };            // groups 2/3 unused (2D)
        i32x8 gz8 = { 0, 0, 0, 0, 0, 0, 0, 0 };
        // clang-23 / therock form: 6 args (extra int32x8 block before cpol).
        __builtin_amdgcn_tensor_load_to_lds(g0, g1, gz4, gz4, gz8, 0);
        __builtin_amdgcn_s_wait_tensorcnt(0);  // s_wait_tensorcnt 0
    }
#else
    for (int j = tid; j < A_ANCH; j += K2_THREADS) vals[j] = src[j];
#endif
    __syncthreads();

    // MSB-first radix select for the K-th largest key.
    uint32_t prefix = 0;
    uint32_t want   = (uint32_t)K;
    for (int shift = 24; shift >= 0; shift -= 8) {
        if (tid < 256) hist[tid] = 0;
        __syncthreads();
        const uint32_t himask = (shift == 24) ? 0u : (0xFFFFFFFFu << (shift + 8));
        for (int j = tid; j < A_ANCH; j += K2_THREADS) {
            uint32_t key = f2key(vals[j]);
            if ((key & himask) == prefix)
                atomicAdd(&hist[(key >> shift) & 255u], 1u);
        }
        __syncthreads();
        if (tid == 0) {
            uint32_t cum = 0;
            int b = 255;
            for (; b > 0; --b) {
                uint32_t c = hist[b];
                if (cum + c >= want) break;
                cum += c;
            }
            sh_prefix = prefix | ((uint32_t)b << shift);
            sh_want   = want - cum;
        }
        __syncthreads();
        prefix = sh_prefix;
        want   = sh_want;
        __syncthreads();
    }

    // Top-K sum = sum(key > T) + (#ties to include) * value(T).
    const uint32_t tkey = prefix;
    const float    tval = key2f(tkey);
    float part = 0.0f;
    for (int j = tid; j < A_ANCH; j += K2_THREADS) {
        float v = vals[j];
        if (f2key(v) > tkey) part += v;
    }
#pragma unroll
    for (int o = 16; o > 0; o >>= 1) part += __shfl_xor(part, o, 32);
    if ((tid & 31) == 0) redbuf[tid >> 5] = part;
    __syncthreads();
    if (tid < 32) {
        float r = (tid < K2_THREADS / 32) ? redbuf[tid] : 0.0f;
#pragma unroll
        for (int o = 16; o > 0; o >>= 1) r += __shfl_xor(r, o, 32);
        if (tid == 0) atomicAdd(&acc[1], r + (float)want * tval);
    }
}

// ---------------------------------------------------------------------------
// Kernel 3: finalize scalars.
__global__ void k_fin(const int* __restrict__ numpos,
                      const float* __restrict__ acc,
                      float* __restrict__ out, int out_size) {
    if (threadIdx.x == 0 && blockIdx.x == 0) {
        int nb = 0;
#pragma unroll
        for (int n = 0; n < N_IMG; ++n) nb += numpos[n];
        const bool  nopos = (nb == 0);
        const float denom = (float)(nb > 0 ? nb : 1);
        if (out_size > 0) out[0] = nopos ? 0.0f : acc[0] / denom;
        if (out_size > 1) out[1] = nopos ? 0.0f : acc[1] / denom;
        if (out_size > 2) out[2] = nopos ? 1.0f : 0.0f;
    }
}

// ---------------------------------------------------------------------------
extern "C" void kernel_launch(void* const* d_in, const int* in_sizes, int n_in,
                              void* d_out, int out_size, void* d_ws, size_t ws_size,
                              hipStream_t stream) {
    const float* regs   = (const float*)d_in[0];
    const float* cls    = (const float*)d_in[1];
    const float* coords = (const float*)d_in[2];
    const int*   labels = (const int*)d_in[3];

    // Workspace layout: [N*A] loss_c floats | [N] num_pos ints | [2] f32 acc.
    float* lossc  = (float*)d_ws;
    int*   numpos = (int*)((char*)d_ws + (size_t)N_IMG * A_ANCH * sizeof(float));
    float* acc    = (float*)(numpos + N_IMG);

    k_init<<<1, 64, 0, stream>>>(numpos, acc);
    k_ce<<<(N_IMG * A_ANCH) / 256, 256, 0, stream>>>(regs, cls, coords, labels,
                                                     lossc, numpos, acc);
    k_topk<<<N_IMG, K2_THREADS, 0, stream>>>(lossc, numpos, acc);
    k_fin<<<1, 1, 0, stream>>>(numpos, acc, (float*)d_out, out_size);
}